// RWKV_RNN_37795712205159
// MI455X (gfx1250) — compile-verified
//
#include <hip/hip_runtime.h>
#include <hip/hip_bf16.h>

// RWKV-v4 single-token step for gfx1250 (MI455X).
// Bandwidth-bound GEMV workload (~1.44 GB weights/token -> ~62us @ 23.3TB/s).
// Matrix math via V_WMMA_F32_16X16X4_F32 (fp32 in/out, 16-row GEMV tiles).

#define EDIM   1024
#define HDIM   4096
#define NLAYER 24
#define VOCAB  32000

typedef __attribute__((ext_vector_type(2))) float v2f;
typedef __attribute__((ext_vector_type(4))) float v4f;
typedef __attribute__((ext_vector_type(8))) float v8f;

__device__ __forceinline__ float sigmoid_(float x) { return 1.f / (1.f + __expf(-x)); }

// ---------------------------------------------------------------------------
// Block-wide LayerNorm over 1024 elements, blockDim.x == 256 (8 waves, wave32).
// ---------------------------------------------------------------------------
__device__ __forceinline__ void block_ln_1024(const float* __restrict__ xg,
                                              const float* __restrict__ gw,
                                              const float* __restrict__ gb,
                                              float* out_lds, float* red) {
  const int tid = threadIdx.x;
  float v0 = xg[tid], v1 = xg[tid + 256], v2 = xg[tid + 512], v3 = xg[tid + 768];
  float s = (v0 + v1) + (v2 + v3);
#pragma unroll
  for (int off = 16; off > 0; off >>= 1) s += __shfl_xor(s, off, 32);
  if ((tid & 31) == 0) red[tid >> 5] = s;
  __syncthreads();
  float tot = 0.f;
#pragma unroll
  for (int i = 0; i < 8; ++i) tot += red[i];
  const float mean = tot * (1.f / 1024.f);
  float d0 = v0 - mean, d1 = v1 - mean, d2 = v2 - mean, d3 = v3 - mean;
  float s2 = d0 * d0 + d1 * d1 + d2 * d2 + d3 * d3;
#pragma unroll
  for (int off = 16; off > 0; off >>= 1) s2 += __shfl_xor(s2, off, 32);
  if ((tid & 31) == 0) red[8 + (tid >> 5)] = s2;
  __syncthreads();
  float tot2 = 0.f;
#pragma unroll
  for (int i = 0; i < 8; ++i) tot2 += red[8 + i];
  const float rstd = rsqrtf(tot2 * (1.f / 1024.f) + 1e-5f);
  out_lds[tid]       = d0 * rstd * gw[tid]       + gb[tid];
  out_lds[tid + 256] = d1 * rstd * gw[tid + 256] + gb[tid + 256];
  out_lds[tid + 512] = d2 * rstd * gw[tid + 512] + gb[tid + 512];
  out_lds[tid + 768] = d3 * rstd * gw[tid + 768] + gb[tid + 768];
  __syncthreads();
}

// ---------------------------------------------------------------------------
// 16-row GEMV partial via V_WMMA_F32_16X16X4_F32, K consumed in blocks of 8
// from a single b128 load per lane per iteration:
//   lanes 0-15  load W[m][k0 .. k0+3]   (tile-K positions 0,1 per WMMA)
//   lanes 16-31 load W[m][k0+4 .. k0+7] (tile-K positions 2,3 per WMMA)
// WMMA#0 uses (x,y) -> logical k {k0,k0+1,k0+4,k0+5};
// WMMA#1 uses (z,w) -> logical k {k0+2,k0+3,k0+6,k0+7}.
// B column 0 (lane 0 = tile-K 0,1; lane 16 = tile-K 2,3) matches that order;
// all other lanes are zeroed so columns N!=0 of D stay exactly 0.
// Result column 0 of D: lane 0 holds M=0..7, lane 16 holds M=8..15.
// ---------------------------------------------------------------------------
__device__ __forceinline__ v8f wmma_gemv16(const float* __restrict__ W, int ldw,
                                           int K, const float* __restrict__ xv,
                                           int row_base) {
  const int lane = threadIdx.x & 31;
  const int half = lane >> 4;  // 0: lanes 0-15, 1: lanes 16-31
  const int m    = row_base + (lane & 15);
  const float* wp = W + (size_t)m * (size_t)ldw + 4 * half;
  const float* bp = xv + 4 * half;
  const float msk = ((lane & 15) == 0) ? 1.f : 0.f;  // keep lane 0 / lane 16
  v8f acc0 = {0.f, 0.f, 0.f, 0.f, 0.f, 0.f, 0.f, 0.f};
  v8f acc1 = {0.f, 0.f, 0.f, 0.f, 0.f, 0.f, 0.f, 0.f};
  for (int k0 = 0; k0 < K; k0 += 8) {
    v4f a = __builtin_nontemporal_load((const v4f*)(wp + k0));
    v2f a0, a1, b0, b1;
    a0.x = a.x; a0.y = a.y;
    a1.x = a.z; a1.y = a.w;
    b0.x = bp[k0 + 0] * msk;
    b0.y = bp[k0 + 1] * msk;
    b1.x = bp[k0 + 2] * msk;
    b1.y = bp[k0 + 3] * msk;
    acc0 = __builtin_amdgcn_wmma_f32_16x16x4_f32(false, a0, false, b0,
                                                 (short)0, acc0, false, false);
    acc1 = __builtin_amdgcn_wmma_f32_16x16x4_f32(false, a1, false, b1,
                                                 (short)0, acc1, false, false);
  }
  return acc0 + acc1;
}

// ---------------------------------------------------------------------------
// Block-cooperative GEMV: 8 waves, SPLIT waves per 16-row tile (K-split for
// more in-flight HBM streams), NT = 8/SPLIT tiles per block.  Partials are
// combined through a 512B LDS buffer, then f(row, value) is applied once per
// output row by the first NT*16 threads.
// ---------------------------------------------------------------------------
template <int SPLIT, int NT, typename F>
__device__ __forceinline__ void gemv_tiles(const float* __restrict__ W, int ldw,
                                           int K, const float* __restrict__ xv,
                                           int block_row_base,
                                           float (*part)[16], F f) {
  const int wid  = threadIdx.x >> 5;
  const int tile = wid / SPLIT;
  const int seg  = wid % SPLIT;
  const int Kseg = K / SPLIT;
  v8f acc = wmma_gemv16(W + (size_t)seg * Kseg, ldw, Kseg, xv + seg * Kseg,
                        block_row_base + tile * 16);
  const int lane = threadIdx.x & 31;
  if ((lane & 15) == 0) {
    float* p = part[wid] + (lane >> 4) * 8;
#pragma unroll
    for (int i = 0; i < 8; ++i) p[i] = acc[i];
  }
  __syncthreads();
  const int tid = threadIdx.x;
  if (tid < NT * 16) {
    const int t = tid >> 4, r = tid & 15;
    float sum = 0.f;
#pragma unroll
    for (int sg = 0; sg < SPLIT; ++sg) sum += part[t * SPLIT + sg][r];
    f(block_row_base + t * 16 + r, sum);
  }
}

// ---------------------------------------------------------------------------
// Kernel 1: x = LN(emb[token]) * ln0_w + ln0_b          <<<1, 256>>>
// ---------------------------------------------------------------------------
__global__ void k_embed_ln0(const int* __restrict__ token,
                            const float* __restrict__ emb,
                            const float* __restrict__ w,
                            const float* __restrict__ b,
                            float* __restrict__ x_out) {
  __shared__ float xl[EDIM];
  __shared__ float red[16];
  block_ln_1024(emb + (size_t)token[0] * EDIM, w, b, xl, red);
  for (int i = threadIdx.x; i < EDIM; i += 256) x_out[i] = xl[i];
}

// ---------------------------------------------------------------------------
// Kernel 2 (per layer): time-mix prolog + k/v/r GEMVs   <<<96, 256>>>
// blocks 0-31 -> att_key, 32-63 -> att_value, 64-95 -> att_rec (sigmoid).
// Each block owns 32 rows (2 tiles x 16), 4-way K-split.
// ---------------------------------------------------------------------------
__global__ void k_tmix_qkv(const float* __restrict__ x,
                           const float* __restrict__ state_l,
                           const float* __restrict__ l1w, const float* __restrict__ l1b,
                           const float* __restrict__ tmk, const float* __restrict__ tmv,
                           const float* __restrict__ tmr,
                           const float* __restrict__ akw, const float* __restrict__ avw,
                           const float* __restrict__ arw,
                           float* __restrict__ k_out, float* __restrict__ v_out,
                           float* __restrict__ r_out, float* __restrict__ xn_state_out) {
  __shared__ float xn[EDIM];
  __shared__ float mix[EDIM];
  __shared__ float red[16];
  __shared__ float part[8][16];
  block_ln_1024(x, l1w, l1b, xn, red);

  const int mat = blockIdx.x >> 5;  // 0,1,2 (block-uniform)
  const int blk = blockIdx.x & 31;
  const float* tm = (mat == 0) ? tmk : (mat == 1) ? tmv : tmr;
  const float* s_att = state_l + EDIM;  // state[1]
  for (int i = threadIdx.x; i < EDIM; i += 256) {
    const float t = tm[i];
    mix[i] = xn[i] * t + s_att[i] * (1.f - t);
  }
  if (blockIdx.x == 0)
    for (int i = threadIdx.x; i < EDIM; i += 256) xn_state_out[i] = xn[i];
  __syncthreads();

  const float* W = (mat == 0) ? akw : (mat == 1) ? avw : arw;
  float* out = (mat == 0) ? k_out : (mat == 1) ? v_out : r_out;
  if (mat == 2)
    gemv_tiles<4, 2>(W, EDIM, EDIM, mix, blk * 32, part,
                     [&](int r, float y) { out[r] = sigmoid_(y); });
  else
    gemv_tiles<4, 2>(W, EDIM, EDIM, mix, blk * 32, part,
                     [&](int r, float y) { out[r] = y; });
}

// ---------------------------------------------------------------------------
// Kernel 3 (per layer): WKV + x += att_out @ (r*wkv) + state update <<<32,256>>>
// ---------------------------------------------------------------------------
__global__ void k_tmix_wkv_out(float* __restrict__ x,
                               const float* __restrict__ state_l,
                               const float* __restrict__ tfirst,
                               const float* __restrict__ tdecay,
                               const float* __restrict__ kv,
                               const float* __restrict__ vv,
                               const float* __restrict__ rv,
                               const float* __restrict__ aow,
                               float* __restrict__ nst /* layer slice of new_state */) {
  __shared__ float rwkv[EDIM];
  __shared__ float part[8][16];
  const float* aa = state_l + 2 * EDIM;
  const float* bb = state_l + 3 * EDIM;
  const float* pp = state_l + 4 * EDIM;
  for (int i = threadIdx.x; i < EDIM; i += 256) {
    const float k = kv[i], v = vv[i];
    const float ww = tfirst[i] + k;
    const float qq = fmaxf(pp[i], ww);
    const float e1 = __expf(pp[i] - qq), e2 = __expf(ww - qq);
    rwkv[i] = rv[i] * ((e1 * aa[i] + e2 * v) / (e1 * bb[i] + e2));
    if (blockIdx.x == 0) {  // new aa/bb/pp state
      const float ww2 = pp[i] + tdecay[i];
      const float qq2 = fmaxf(ww2, k);
      const float f1 = __expf(ww2 - qq2), f2 = __expf(k - qq2);
      nst[2 * EDIM + i] = f1 * aa[i] + f2 * v;
      nst[3 * EDIM + i] = f1 * bb[i] + f2;
      nst[4 * EDIM + i] = qq2;
    }
  }
  __syncthreads();
  gemv_tiles<4, 2>(aow, EDIM, EDIM, rwkv, blockIdx.x * 32, part,
                   [&](int r, float y) { x[r] += y; });
}

// ---------------------------------------------------------------------------
// Kernel 4 (per layer): channel-mix prolog + ffn_key/ffn_rec GEMVs <<<160,256>>>
// blocks 0-127 -> ffn_key (4096 rows, relu^2); 128-159 -> ffn_rec (sigmoid)
// ---------------------------------------------------------------------------
__global__ void k_ffn_kr(const float* __restrict__ x,
                         const float* __restrict__ state_l,
                         const float* __restrict__ l2w, const float* __restrict__ l2b,
                         const float* __restrict__ ftmk, const float* __restrict__ ftmr,
                         const float* __restrict__ fkw, const float* __restrict__ frw,
                         float* __restrict__ kk_out, float* __restrict__ r2_out,
                         float* __restrict__ xn2_state_out) {
  __shared__ float xn2[EDIM];
  __shared__ float mix[EDIM];
  __shared__ float red[16];
  __shared__ float part[8][16];
  block_ln_1024(x, l2w, l2b, xn2, red);

  const bool isK = blockIdx.x < 128;  // block-uniform
  const float* tm = isK ? ftmk : ftmr;
  const float* s_ffn = state_l;  // state[0]
  for (int i = threadIdx.x; i < EDIM; i += 256) {
    const float t = tm[i];
    mix[i] = xn2[i] * t + s_ffn[i] * (1.f - t);
  }
  if (blockIdx.x == 0)
    for (int i = threadIdx.x; i < EDIM; i += 256) xn2_state_out[i] = xn2[i];
  __syncthreads();

  if (isK)
    gemv_tiles<4, 2>(fkw, EDIM, EDIM, mix, blockIdx.x * 32, part,
                     [&](int r, float y) {
                       const float rl = fmaxf(y, 0.f);
                       kk_out[r] = rl * rl;
                     });
  else
    gemv_tiles<4, 2>(frw, EDIM, EDIM, mix, (blockIdx.x - 128) * 32, part,
                     [&](int r, float y) { r2_out[r] = sigmoid_(y); });
}

// ---------------------------------------------------------------------------
// Kernel 5 (per layer): x += r2 * (ffn_value @ kk), K = 4096     <<<32, 256>>>
// ---------------------------------------------------------------------------
__global__ void k_ffn_out(float* __restrict__ x,
                          const float* __restrict__ kk,
                          const float* __restrict__ r2,
                          const float* __restrict__ fvw) {
  __shared__ float kl[HDIM];  // 16 KB of the 320 KB/WGP LDS
  __shared__ float part[8][16];
  for (int i = threadIdx.x; i < HDIM; i += 256) kl[i] = kk[i];
  __syncthreads();
  gemv_tiles<4, 2>(fvw, HDIM, HDIM, kl, blockIdx.x * 32, part,
                   [&](int r, float y) { x[r] += r2[r] * y; });
}

// ---------------------------------------------------------------------------
// Kernel 6: logits = head @ LN(x)                           <<<250, 256>>>
// 2000 row-tiles already give ample parallelism -> SPLIT=1, 128 rows/block.
// ---------------------------------------------------------------------------
__global__ void k_head(const float* __restrict__ x,
                       const float* __restrict__ w, const float* __restrict__ b,
                       const float* __restrict__ head,
                       float* __restrict__ logits) {
  __shared__ float xo[EDIM];
  __shared__ float red[16];
  __shared__ float part[8][16];
  block_ln_1024(x, w, b, xo, red);
  gemv_tiles<1, 8>(head, EDIM, EDIM, xo, blockIdx.x * 128, part,
                   [&](int r, float y) { logits[r] = y; });
}

// ---------------------------------------------------------------------------
extern "C" void kernel_launch(void* const* d_in, const int* in_sizes, int n_in,
                              void* d_out, int out_size, void* d_ws, size_t ws_size,
                              hipStream_t stream) {
  (void)in_sizes; (void)n_in; (void)out_size; (void)ws_size;
  const int*   token   = (const int*)  d_in[0];
  const float* state   = (const float*)d_in[1];
  const float* emb     = (const float*)d_in[2];
  const float* ln0w    = (const float*)d_in[3];
  const float* ln0b    = (const float*)d_in[4];
  const float* att_key = (const float*)d_in[5];
  const float* att_val = (const float*)d_in[6];
  const float* att_rec = (const float*)d_in[7];
  const float* att_out = (const float*)d_in[8];
  const float* att_tmk = (const float*)d_in[9];
  const float* att_tmv = (const float*)d_in[10];
  const float* att_tmr = (const float*)d_in[11];
  const float* att_tf  = (const float*)d_in[12];
  const float* att_td  = (const float*)d_in[13];
  const float* ln1w    = (const float*)d_in[14];
  const float* ln1b    = (const float*)d_in[15];
  const float* ln2w    = (const float*)d_in[16];
  const float* ln2b    = (const float*)d_in[17];
  const float* ffn_key = (const float*)d_in[18];
  const float* ffn_val = (const float*)d_in[19];
  const float* ffn_rec = (const float*)d_in[20];
  const float* ffn_tmk = (const float*)d_in[21];
  const float* ffn_tmr = (const float*)d_in[22];
  const float* headw   = (const float*)d_in[23];
  const float* lnoutw  = (const float*)d_in[24];
  const float* lnoutb  = (const float*)d_in[25];

  float* out    = (float*)d_out;
  float* logits = out;                 // [32000]
  float* nstate = out + VOCAB;         // [24][5][1024]

  float* ws   = (float*)d_ws;
  float* x    = ws;                    // [1024]
  float* kbuf = ws + 1 * EDIM;         // [1024]
  float* vbuf = ws + 2 * EDIM;         // [1024]
  float* rbuf = ws + 3 * EDIM;         // [1024]
  float* kk   = ws + 4 * EDIM;         // [4096]
  float* r2   = ws + 4 * EDIM + HDIM;  // [1024]

  k_embed_ln0<<<1, 256, 0, stream>>>(token, emb, ln0w, ln0b, x);

  for (int l = 0; l < NLAYER; ++l) {
    const size_t vE  = (size_t)l * EDIM;
    const size_t vEE = (size_t)l * EDIM * EDIM;
    const size_t vHE = (size_t)l * (size_t)HDIM * EDIM;
    const float* st  = state  + (size_t)l * 5 * EDIM;
    float*       nst = nstate + (size_t)l * 5 * EDIM;

    k_tmix_qkv<<<96, 256, 0, stream>>>(
        x, st, ln1w + vE, ln1b + vE,
        att_tmk + vE, att_tmv + vE, att_tmr + vE,
        att_key + vEE, att_val + vEE, att_rec + vEE,
        kbuf, vbuf, rbuf, nst + EDIM /* state[1] = xn */);

    k_tmix_wkv_out<<<32, 256, 0, stream>>>(
        x, st, att_tf + vE, att_td + vE, kbuf, vbuf, rbuf,
        att_out + vEE, nst);

    k_ffn_kr<<<160, 256, 0, stream>>>(
        x, st, ln2w + vE, ln2b + vE, ffn_tmk + vE, ffn_tmr + vE,
        ffn_key + vHE, ffn_rec + vEE, kk, r2, nst /* state[0] = xn2 */);

    k_ffn_out<<<32, 256, 0, stream>>>(x, kk, r2, ffn_val + vHE);
  }

  k_head<<<250, 256, 0, stream>>>(x, lnoutw, lnoutb, headw, logits);
}